// Encoder_12463995093415
// MI455X (gfx1250) — compile-verified
//
#include <hip/hip_runtime.h>

typedef __attribute__((ext_vector_type(2))) float v2f;
typedef __attribute__((ext_vector_type(8))) float v8f;

// ---------------- utility kernels ----------------

__global__ void fill_kernel(float* __restrict__ p, float v, int n) {
    int i = blockIdx.x * blockDim.x + threadIdx.x;
    if (i < n) p[i] = v;
}

// deg[dst[e]] += 1  (deg pre-initialized to 1.0 for the self loop)
__global__ void deg_count_kernel(const int* __restrict__ dst,
                                 float* __restrict__ deg, int E) {
    int e = blockIdx.x * blockDim.x + threadIdx.x;
    if (e < E) atomicAdd(&deg[dst[e]], 1.0f);
}

__global__ void rsqrt_kernel(float* __restrict__ d, int n) {
    int i = blockIdx.x * blockDim.x + threadIdx.x;
    if (i < n) d[i] = rsqrtf(d[i]);   // deg >= 1 always (self loop)
}

// ---------------- fp32 WMMA GEMM: Y[nrows,64] = X[nrows,64] @ W[64,64] ----------------
// One wave computes a 16x64 output stripe using V_WMMA_F32_16X16X4_F32.
// Block = 256 threads = 8 waves = 128 rows.
// OOB rows: A-row m only feeds D-row m, so we clamp the A load address and
// simply skip OOB stores — no masking, no divergent loads.

__global__ void __launch_bounds__(256)
gemm64_wmma(const float* __restrict__ X, const float* __restrict__ W,
            float* __restrict__ Y, int nrows) {
    // W staged in LDS, k-pair interleaved so each B fragment is contiguous 8B:
    // lw[(k>>1)*128 + n*2 + (k&1)] = W[k*64 + n]
    __shared__ float lw[64 * 64];
    for (int i = threadIdx.x; i < 64 * 64; i += 256) {
        int k = i >> 6, n = i & 63;
        lw[(k >> 1) * 128 + n * 2 + (k & 1)] = W[i];
    }
    __syncthreads();

    int wave = threadIdx.x >> 5;
    int lane = threadIdx.x & 31;
    int lh   = lane & 15;   // lane % 16
    int hi   = lane >> 4;   // 0 or 1
    int rowbase = (blockIdx.x * 8 + wave) * 16;

    int arow = rowbase + lh;
    int arow_c = arow < nrows ? arow : (nrows - 1);        // clamped, always valid
    const v2f* xp = (const v2f*)(X + (size_t)arow_c * 64); // a(kk) = xp[2*kk+hi]

    v8f acc0 = {}, acc1 = {}, acc2 = {}, acc3 = {};

    #pragma unroll
    for (int kk = 0; kk < 16; ++kk) {
        v2f a = xp[2 * kk + hi];                 // K = 4*kk + 2*hi (+1)

        // B fragments: pair index p = 2*kk + hi, contiguous v2f in LDS
        const v2f* wp = (const v2f*)(lw + (2 * kk + hi) * 128 + lh * 2);
        v2f b0  = wp[0];    // n = lh
        v2f b1v = wp[16];   // n = lh + 16
        v2f b2v = wp[32];   // n = lh + 32
        v2f b3v = wp[48];   // n = lh + 48

        acc0 = __builtin_amdgcn_wmma_f32_16x16x4_f32(false, a, false, b0,  (short)0, acc0, false, false);
        acc1 = __builtin_amdgcn_wmma_f32_16x16x4_f32(false, a, false, b1v, (short)0, acc1, false, false);
        acc2 = __builtin_amdgcn_wmma_f32_16x16x4_f32(false, a, false, b2v, (short)0, acc2, false, false);
        acc3 = __builtin_amdgcn_wmma_f32_16x16x4_f32(false, a, false, b3v, (short)0, acc3, false, false);
    }

    #pragma unroll
    for (int r = 0; r < 8; ++r) {
        int row = rowbase + r + hi * 8;   // C/D layout: M = r + 8*hi
        if (row < nrows) {
            float* yp = Y + (size_t)row * 64 + lh;
            yp[0]  = acc0[r];
            yp[16] = acc1[r];
            yp[32] = acc2[r];
            yp[48] = acc3[r];
        }
    }
}

// ---------------- edge scatter: agg[dst] += xw[src] * dinv[src]*dinv[dst] ----------------
// 16 threads per edge; each thread gathers a float4 and issues 4 f32 atomics.

__global__ void scatter_kernel(const float* __restrict__ xw,
                               const int* __restrict__ src,
                               const int* __restrict__ dst,
                               const float* __restrict__ dinv,
                               float* __restrict__ agg, int E) {
    int t = blockIdx.x * blockDim.x + threadIdx.x;
    int e = t >> 4;
    int q = t & 15;
    if (e >= E) return;
    int s = src[e];
    int d = dst[e];
    float nrm = dinv[s] * dinv[d];
    float4 v = reinterpret_cast<const float4*>(xw + (size_t)s * 64)[q];
    float* base = agg + (size_t)d * 64 + q * 4;
    atomicAdd(base + 0, v.x * nrm);
    atomicAdd(base + 1, v.y * nrm);
    atomicAdd(base + 2, v.z * nrm);
    atomicAdd(base + 3, v.w * nrm);
}

// ---------------- epilogue: acc += xw * dinv^2 (self loop) + bias ; optional ReLU ----------------

__global__ void finish_kernel(float* __restrict__ acc, const float* __restrict__ xw,
                              const float* __restrict__ dinv, const float* __restrict__ bias,
                              int total, int do_relu) {
    int i = blockIdx.x * blockDim.x + threadIdx.x;
    if (i >= total) return;
    int n = i >> 6;
    int f = i & 63;
    float di = dinv[n];
    float v = acc[i] + xw[i] * di * di + bias[f];
    if (do_relu) v = fmaxf(v, 0.0f);
    acc[i] = v;
}

// ---------------- launch ----------------

extern "C" void kernel_launch(void* const* d_in, const int* in_sizes, int n_in,
                              void* d_out, int out_size, void* d_ws, size_t ws_size,
                              hipStream_t stream) {
    (void)n_in; (void)out_size; (void)ws_size;

    const float* x  = (const float*)d_in[0];
    const int*   ei = (const int*)d_in[1];
    const float* W1 = (const float*)d_in[2];
    const float* b1 = (const float*)d_in[3];
    const float* W2 = (const float*)d_in[4];
    const float* b2 = (const float*)d_in[5];
    float* out = (float*)d_out;

    const int N = in_sizes[0] / 64;      // 100000
    const int E = in_sizes[1] / 2;       // 1200000
    const int* src = ei;
    const int* dst = ei + E;

    float* dinv = (float*)d_ws;
    float* xw   = dinv + (((size_t)N + 32767) & ~(size_t)32767); // 128KB-aligned region
    float* agg  = xw + (size_t)N * 64;

    const int T = 256;
    const int NF = N * 64;

    // --- degree / normalization ---
    fill_kernel<<<(N + T - 1) / T, T, 0, stream>>>(dinv, 1.0f, N);
    deg_count_kernel<<<(E + T - 1) / T, T, 0, stream>>>(dst, dinv, E);
    rsqrt_kernel<<<(N + T - 1) / T, T, 0, stream>>>(dinv, N);

    // --- layer 1: h = relu(Anorm @ (x@W1) + b1) ---
    gemm64_wmma<<<(N + 127) / 128, T, 0, stream>>>(x, W1, xw, N);
    fill_kernel<<<(NF + T - 1) / T, T, 0, stream>>>(agg, 0.0f, NF);
    scatter_kernel<<<((E * 16) + T - 1) / T, T, 0, stream>>>(xw, src, dst, dinv, agg, E);
    finish_kernel<<<(NF + T - 1) / T, T, 0, stream>>>(agg, xw, dinv, b1, NF, 1);

    // --- layer 2: out = Anorm @ (h@W2) + b2 ---
    gemm64_wmma<<<(N + 127) / 128, T, 0, stream>>>(agg, W2, xw, N);
    fill_kernel<<<(NF + T - 1) / T, T, 0, stream>>>(out, 0.0f, NF);
    scatter_kernel<<<((E * 16) + T - 1) / T, T, 0, stream>>>(xw, src, dst, dinv, out, E);
    finish_kernel<<<(NF + T - 1) / T, T, 0, stream>>>(out, xw, dinv, b2, NF, 0);
}